// PointCapsNetSeg2Loss_60138132078931
// MI455X (gfx1250) — compile-verified
//
#include <hip/hip_runtime.h>

typedef __attribute__((ext_vector_type(2))) float v2f;
typedef __attribute__((ext_vector_type(8))) float v8f;

__global__ __launch_bounds__(32) void zero_out_kernel(float* out) {
    if (threadIdx.x == 0) out[0] = 0.0f;
}

// Computes  weight * sum_over_rows( min_over_cols ||x_i - y_j||^2 )
// for a set of sub-batches, accumulated into *out with atomicAdd.
//
// Point layout: coord c of point i of sub-batch k lives at
//   base + (k/nparts)*outerStride + (k%nparts)*N + c*coordStride + i
// (matches the [B,3,N] / per-part-sliced layouts of the reference).
//
// Trick: A row = (x, y, z, 1), B col = (-2x', -2y', -2z', ||y||^2)
// so the 16x16x4 f32 WMMA directly yields D = dist(x,y) - ||x||^2.
// The -||x||^2 shift is constant per row -> row-min ordering preserved;
// Sum(||x||^2) is added back from the A-tile registers.
__global__ __launch_bounds__(32) void chamfer_min_sum_kernel(
    const float* __restrict__ X, const float* __restrict__ Y,
    int NX, int NY, int nparts,
    int xOuter, int xCoord, int yOuter, int yCoord,
    float weight, float* __restrict__ out)
{
    const int lane = threadIdx.x & 31;
    const int half = lane >> 4;      // 0: K=0,1 rows/cols ; 1: K=2,3
    const int l    = lane & 15;

    const int k = blockIdx.y;
    const int b = k / nparts;
    const int p = k - b * nparts;
    const float* xb = X + b * xOuter + p * NX;
    const float* yb = Y + b * yOuter + p * NY;

    // ---- A tile: 16 rows (points mi), K = (x, y, z, 1) ----
    const int mi = blockIdx.x * 16 + l;
    const float ax = xb[0 * xCoord + mi];
    const float ay = xb[1 * xCoord + mi];
    const float az = xb[2 * xCoord + mi];

    v2f A;
    A.x = half ? az   : ax;   // VGPR0: K=0 (lanes 0-15) / K=2 (lanes 16-31)
    A.y = half ? 1.0f : ay;   // VGPR1: K=1 / K=3
    // per-lane piece of Sum_i ||x_i||^2 over the 16-row strip
    const float nasq = half ? (az * az) : (ax * ax + ay * ay);

    v8f acc;
#pragma unroll
    for (int i = 0; i < 8; ++i) acc[i] = 3.0e38f;

    for (int j0 = 0; j0 < NY; j0 += 16) {
        const float bx = yb[0 * yCoord + j0 + l];
        const float by = yb[1 * yCoord + j0 + l];
        const float bz = yb[2 * yCoord + j0 + l];
        v2f Bv;
        Bv.x = half ? (-2.0f * bz) : (-2.0f * bx);               // K=0 / K=2
        Bv.y = half ? (bx * bx + by * by + bz * bz)              // K=3: ||y||^2
                    : (-2.0f * by);                              // K=1
        v8f c = {};
        // D = A x B + 0 : 16x16 tile of (dist - ||x||^2)
        v8f d = __builtin_amdgcn_wmma_f32_16x16x4_f32(
            false, A, false, Bv, (short)0, c, false, false);
#pragma unroll
        for (int i = 0; i < 8; ++i) acc[i] = fminf(acc[i], d[i]);
    }

    // Row mins: C/D layout puts row M = v + 8*half in VGPR v across the
    // 16 lanes of a half-wave -> xor-min over masks 1,2,4,8.
#pragma unroll
    for (int m = 1; m <= 8; m <<= 1) {
#pragma unroll
        for (int i = 0; i < 8; ++i)
            acc[i] = fminf(acc[i], __shfl_xor(acc[i], m, 32));
    }
    float rsum = 0.0f;          // lane l==0 of each half: Sum of 8 row-mins
#pragma unroll
    for (int i = 0; i < 8; ++i) rsum += acc[i];

    // strip total = Sum(row mins, shifted) + Sum ||x||^2
    float contrib = nasq + ((l == 0) ? rsum : 0.0f);
#pragma unroll
    for (int m = 1; m <= 16; m <<= 1) contrib += __shfl_xor(contrib, m, 32);

    if (lane == 0) atomicAdd(out, contrib * weight);
}

extern "C" void kernel_launch(void* const* d_in, const int* in_sizes, int n_in,
                              void* d_out, int out_size, void* d_ws, size_t ws_size,
                              hipStream_t stream) {
    (void)in_sizes; (void)n_in; (void)out_size; (void)d_ws; (void)ws_size;
    const float* data = (const float*)d_in[0];  // [8, 3, 8192]
    const float* rec  = (const float*)d_in[1];  // [8, 3, 2048]
    const float* recp = (const float*)d_in[2];  // [8, 3, 8192]
    float* out = (float*)d_out;

    const int B = 8, N = 2048, P = 4;
    const int NP = N * P;  // 8192

    zero_out_kernel<<<dim3(1), dim3(32), 0, stream>>>(out);

    dim3 blk(32);

    // mean(d1): rows = data (8192), cols = rec (2048)
    chamfer_min_sum_kernel<<<dim3(NP / 16, B), blk, 0, stream>>>(
        data, rec, NP, N, 1, 3 * NP, NP, 3 * N, N,
        1.0f / (5.0f * B * NP), out);

    // mean(d2): rows = rec (2048), cols = data (8192)
    chamfer_min_sum_kernel<<<dim3(N / 16, B), blk, 0, stream>>>(
        rec, data, N, NP, 1, 3 * N, N, 3 * NP, NP,
        1.0f / (5.0f * B * N), out);

    // per-part p1: rows = data slice, cols = recp slice (32 sub-batches)
    chamfer_min_sum_kernel<<<dim3(N / 16, B * P), blk, 0, stream>>>(
        data, recp, N, N, P, 3 * NP, NP, 3 * NP, NP,
        1.0f / (5.0f * B * N), out);

    // per-part p2: rows = recp slice, cols = data slice
    chamfer_min_sum_kernel<<<dim3(N / 16, B * P), blk, 0, stream>>>(
        recp, data, N, N, P, 3 * NP, NP, 3 * NP, NP,
        1.0f / (5.0f * B * N), out);
}